// NCC_42537356100315
// MI455X (gfx1250) — compile-verified
//
#include <hip/hip_runtime.h>
#include <hip/hip_bf16.h>
#include <math.h>

// ---------------------------------------------------------------------------
// NCC on gfx1250 (MI455X):
//   res  = depthwise corr(inputs, znorm(template)/441), VALID 21x21
//   out  = res / (sqrt(boxmean(x^2) - boxmean(x)^2 + eps) + eps)
//
// Correlation runs on V_WMMA_F32_16X16X4_F32 via a Toeplitz-style implicit
// GEMM. Each wave computes a 16x32 output strip as TWO chained GEMMs
// D(16x16) = A(16x756) x B(756x16) that SHARE the B fragment (template
// taps), so the steady state is 1 DS instruction per WMMA:
//   ds_load_2addr_b64  -> A fragments for both tiles (same row, +64B)
//   ds_load_2addr_b32  -> shared B fragment
// ---------------------------------------------------------------------------

typedef __attribute__((ext_vector_type(2))) float v2f;
typedef __attribute__((ext_vector_type(8))) float v8f;

#define NCC_EPS   1e-8f
#define TPAD_W    52      // padded template row: idx = (xc - n) + 15 in [0,50]
#define IMG_W     512
#define OUT_W     492
#define PSTR      52      // patch row stride (16+16+20 columns)
#define PATCH_H   36

// ---------------- template z-normalization (one wave per (b,c)) ------------
__global__ __launch_bounds__(32) void ncc_tnorm(const float* __restrict__ t,
                                                float* __restrict__ nt) {
    const int bc = blockIdx.x;           // 48 = B*C
    const int lane = threadIdx.x;        // wave32
    const float* tp = t + bc * 441;

    float s = 0.f, s2 = 0.f;
    for (int i = lane; i < 441; i += 32) {
        float v = tp[i];
        s += v;
        s2 = fmaf(v, v, s2);
    }
    for (int off = 16; off > 0; off >>= 1) {
        s  += __shfl_down(s, off, 32);
        s2 += __shfl_down(s2, off, 32);
    }
    s  = __shfl(s, 0, 32);
    s2 = __shfl(s2, 0, 32);

    const float inv = 1.f / 441.f;
    float mean = s * inv;
    float var  = fmaxf(s2 * inv - mean * mean, 0.f);
    float stdv = sqrtf(var);
    float scale = 1.f / ((stdv + NCC_EPS) * 441.f);   // znorm then /(th*tw)

    float* np = nt + bc * 441;
    for (int i = lane; i < 441; i += 32)
        np[i] = (tp[i] - mean) * scale;
}

// ---------------- main WMMA kernel -----------------------------------------
__global__ __launch_bounds__(128) void ncc_main(const float* __restrict__ inp,
                                                const float* __restrict__ nt,
                                                float* __restrict__ out) {
    __shared__ float tpad[21 * TPAD_W];                   // padded template
    __shared__ alignas(16) float patch[4][PATCH_H * PSTR]; // per-wave 36x52
    __shared__ float colB[4][16 * PSTR];                  // per-wave col sums
    __shared__ float recipT[4][512];                      // per-wave 16x32

    const int bc   = blockIdx.y;          // 0..47
    const int y0   = blockIdx.x * 16;     // tile row origin (0..480)
    const int tid  = threadIdx.x;
    const int wave = tid >> 5;
    const int lane = tid & 31;
    const int lane15 = lane & 15;
    const int hl     = lane >> 4;         // half-wave select (K pairing)

    // Build zero-padded template rows once per block.
    const float* ntp = nt + bc * 441;
    for (int i = tid; i < 21 * TPAD_W; i += 128) {
        int ky  = i / TPAD_W;
        int ii  = i - ky * TPAD_W;
        int idx = ii - 15;                // tap index (xc - n)
        tpad[i] = (idx >= 0 && idx <= 20) ? ntp[ky * 21 + idx] : 0.f;
    }
    __syncthreads();

    float* p  = patch[wave];
    float* cb = colB[wave];
    float* rt = recipT[wave];
    const float* src0 = inp + (size_t)bc * (IMG_W * IMG_W);
    float*       dst0 = out + (size_t)bc * (OUT_W * OUT_W);

    // 16 strips of 32 output columns; 4 per wave (uniform trip count).
    for (int sx = wave; sx < 16; sx += 4) {
        const int x0 = sx * 32;

        // ---- stage 1: load 36x52 patch into LDS (zero-pad at image edge) --
        for (int r = 0; r < PATCH_H; ++r) {
            const int gy = y0 + r;
            const bool rok = (gy < IMG_W);
            const float* srow = src0 + gy * IMG_W;
            const int gx = x0 + lane;
            p[r * PSTR + lane] = (rok && gx < IMG_W) ? srow[gx] : 0.f;
            if (lane < 20) {
                const int gx2 = gx + 32;
                p[r * PSTR + lane + 32] =
                    (rok && gx2 < IMG_W) ? srow[gx2] : 0.f;
            }
        }
        asm volatile("s_wait_dscnt 0x0" ::: "memory");

        // ---- stage 2: dual-tile correlation, shared B fragment ------------
        v8f acc0 = {0.f, 0.f, 0.f, 0.f, 0.f, 0.f, 0.f, 0.f};
        v8f acc1 = {0.f, 0.f, 0.f, 0.f, 0.f, 0.f, 0.f, 0.f};
        for (int ky = 0; ky < 21; ++ky) {
            const float* Arow = p + (lane15 + ky) * PSTR + 2 * hl;
            const float* Brow = tpad + ky * TPAD_W + 15 + 2 * hl - lane15;
#pragma unroll
            for (int q = 0; q < 9; ++q) {          // xc = 4q .. 4q+3
                v2f a0 = *(const v2f*)(Arow + 4 * q);        // tile 0
                v2f a1 = *(const v2f*)(Arow + 4 * q + 16);   // tile 1 (+64B)
                v2f b;
                b.x = Brow[4 * q];
                b.y = Brow[4 * q + 1];
                acc0 = __builtin_amdgcn_wmma_f32_16x16x4_f32(
                        false, a0, false, b, (short)0, acc0, false, false);
                acc1 = __builtin_amdgcn_wmma_f32_16x16x4_f32(
                        false, a1, false, b, (short)0, acc1, false, false);
            }
        }

        // ---- stage 3a: box SUM via separable sliding sums -----------------
        for (int xx = lane; xx < PSTR; xx += 32) {
            float s = 0.f;
#pragma unroll
            for (int r = 0; r < 21; ++r) s += p[r * PSTR + xx];
            cb[xx] = s;
            for (int m = 1; m < 16; ++m) {
                s += p[(m + 20) * PSTR + xx] - p[(m - 1) * PSTR + xx];
                cb[m * PSTR + xx] = s;
            }
        }
        asm volatile("s_wait_dscnt 0x0" ::: "memory");
        {   // row pass: lane -> (m = lane>>1, n in [n0, n0+16))
            const int m  = lane >> 1;
            const int n0 = (lane & 1) * 16;
            float s = 0.f;
#pragma unroll
            for (int kx = 0; kx < 21; ++kx) s += cb[m * PSTR + n0 + kx];
            rt[m * 32 + n0] = s;                       // park raw box sum
            for (int j = 1; j < 16; ++j) {
                const int n = n0 + j;
                s += cb[m * PSTR + n + 20] - cb[m * PSTR + n - 1];
                rt[m * 32 + n] = s;
            }
        }
        asm volatile("s_wait_dscnt 0x0" ::: "memory");

        // ---- stage 3b: box SUMSQ pass, then combine into 1/(std+eps) ------
        for (int xx = lane; xx < PSTR; xx += 32) {
            float s2 = 0.f;
#pragma unroll
            for (int r = 0; r < 21; ++r) {
                float v = p[r * PSTR + xx];
                s2 = fmaf(v, v, s2);
            }
            cb[xx] = s2;
            for (int m = 1; m < 16; ++m) {
                float vi = p[(m + 20) * PSTR + xx];
                float vo = p[(m - 1) * PSTR + xx];
                s2 += vi * vi - vo * vo;
                cb[m * PSTR + xx] = s2;
            }
        }
        asm volatile("s_wait_dscnt 0x0" ::: "memory");
        {
            const int m  = lane >> 1;
            const int n0 = (lane & 1) * 16;
            const float inv = 1.f / 441.f;
            float s2 = 0.f;
#pragma unroll
            for (int kx = 0; kx < 21; ++kx) s2 += cb[m * PSTR + n0 + kx];
            {
                float mean = rt[m * 32 + n0] * inv, msq = s2 * inv;
                float stdv = sqrtf(msq - mean * mean + NCC_EPS);
                rt[m * 32 + n0] = 1.f / (stdv + NCC_EPS);  // same-lane slot
            }
            for (int j = 1; j < 16; ++j) {
                const int n = n0 + j;
                s2 += cb[m * PSTR + n + 20] - cb[m * PSTR + n - 1];
                float mean = rt[m * 32 + n] * inv, msq = s2 * inv;
                float stdv = sqrtf(msq - mean * mean + NCC_EPS);
                rt[m * 32 + n] = 1.f / (stdv + NCC_EPS);
            }
        }
        asm volatile("s_wait_dscnt 0x0" ::: "memory");

        // ---- stage 4: scale by 1/(std+eps) and store (edge-guarded) -------
        // D layout: VGPR r holds M = r + 8*hl, N = lane15.
#pragma unroll
        for (int r = 0; r < 8; ++r) {
            const int m   = r + 8 * hl;
            const int gy  = y0 + m;
            const int gx0 = x0 + lane15;
            const int gx1 = gx0 + 16;
            if (gy < OUT_W && gx0 < OUT_W)
                dst0[gy * OUT_W + gx0] = acc0[r] * rt[m * 32 + lane15];
            if (gy < OUT_W && gx1 < OUT_W)
                dst0[gy * OUT_W + gx1] = acc1[r] * rt[m * 32 + 16 + lane15];
        }
        asm volatile("s_wait_dscnt 0x0" ::: "memory");  // before patch reuse
    }
}

// ---------------------------------------------------------------------------
extern "C" void kernel_launch(void* const* d_in, const int* in_sizes, int n_in,
                              void* d_out, int out_size, void* d_ws, size_t ws_size,
                              hipStream_t stream) {
    const float* inp  = (const float*)d_in[0];   // (16,3,512,512) f32
    const float* tmpl = (const float*)d_in[1];   // (16,3,21,21)  f32
    float* out = (float*)d_out;                  // (16,3,492,492) f32
    float* nt  = (float*)d_ws;                   // 48*441 floats scratch

    ncc_tnorm<<<48, 32, 0, stream>>>(tmpl, nt);

    dim3 grid(31, 48);                           // 31 tile-rows x (B*C)
    ncc_main<<<grid, 128, 0, stream>>>(inp, nt, out);
}